// RelativeGeoSymLoss_88390426951953
// MI455X (gfx1250) — compile-verified
//
#include <hip/hip_runtime.h>
#include <hip/hip_bf16.h>
#include <math.h>

typedef float v2f __attribute__((ext_vector_type(2)));
typedef float v8f __attribute__((ext_vector_type(8)));

// ---------------- workspace layout (float offsets) ----------------
// [0,72)       relPredRot   8*9
// [72,96)      relPredTrans 8*3
// [96,168)     relGtRot     8*9
// [168,192)    relGtTrans   8*3
// [192,4288)   partials     2 * 8 * 256   (pass, batch, rowtile)
// [4288,135360) Pb: pred pts, WMMA-B-swizzled float2  [b][tile][32 lanes]
//                half0 lane n = (x, z)  (K0,K2 rows)
//                half1 lane n = (y, |p|^2) (K1,K3 rows)
// [69824,200896) Gb: gt pts, same layout
#define WS_RELP_R   0
#define WS_RELP_T   72
#define WS_RELG_R   96
#define WS_RELG_T   168
#define WS_PART     192
#define WS_PB       4288
#define WS_GB       69824

__device__ __forceinline__ void rel_xform(const float* Ra, const float* ta,
                                          const float* Rb, const float* tb,
                                          float* Rrel, float* trel) {
  // Rrel = Ra @ Rb^T ; trel = ta - Rrel @ tb
  #pragma unroll
  for (int i = 0; i < 3; ++i)
    #pragma unroll
    for (int k = 0; k < 3; ++k) {
      float s = 0.f;
      #pragma unroll
      for (int j = 0; j < 3; ++j) s += Ra[i*3+j] * Rb[k*3+j];
      Rrel[i*3+k] = s;
    }
  #pragma unroll
  for (int i = 0; i < 3; ++i) {
    float s = ta[i];
    #pragma unroll
    for (int j = 0; j < 3; ++j) s -= Rrel[i*3+j] * tb[j];
    trel[i] = s;
  }
}

__global__ void setup_kernel(const float* __restrict__ pr, const float* __restrict__ pt,
                             const float* __restrict__ chr, const float* __restrict__ cht,
                             const float* __restrict__ gr, const float* __restrict__ gt,
                             const float* __restrict__ cgr, const float* __restrict__ cgt,
                             float* __restrict__ ws) {
  int b = threadIdx.x;
  if (b < 8) {
    rel_xform(pr + 9*b, pt + 3*b, chr + 9*b, cht + 3*b,
              ws + WS_RELP_R + 9*b, ws + WS_RELP_T + 3*b);
    rel_xform(gr + 9*b, gt + 3*b, cgr + 9*b, cgt + 3*b,
              ws + WS_RELG_R + 9*b, ws + WS_RELG_T + 3*b);
  }
}

// Transform each model point by both relative transforms and scatter it
// directly into the WMMA B-operand lane layout so the chamfer pass can load
// B registers with a single b64 load per lane and ZERO swizzle VALU:
//   tile t, half0 lane n : (x_n,  z_n)      -> B rows K0,K2
//   tile t, half1 lane n : (y_n, |v|^2_n)   -> B rows K1,K3
__global__ __launch_bounds__(256)
void transform_kernel(const float* __restrict__ mp, const float* __restrict__ ws,
                      float2* __restrict__ Pb, float2* __restrict__ Gb) {
  int idx = blockIdx.x * 256 + threadIdx.x;      // 0 .. 32767  (b*4096 + n)
  int b = idx >> 12;
  int n = idx & 4095;
  int base = ((b << 8) + (n >> 4)) << 5;         // (b*256 + tile) * 32
  int m = n & 15;

  const float* R0 = ws + WS_RELP_R + 9*b;
  const float* t0 = ws + WS_RELP_T + 3*b;
  const float* R1 = ws + WS_RELG_R + 9*b;
  const float* t1 = ws + WS_RELG_T + 3*b;
  float x = mp[idx*3 + 0], y = mp[idx*3 + 1], z = mp[idx*3 + 2];

  float px = R0[0]*x + R0[1]*y + R0[2]*z + t0[0];
  float py = R0[3]*x + R0[4]*y + R0[5]*z + t0[1];
  float pz = R0[6]*x + R0[7]*y + R0[8]*z + t0[2];
  Pb[base + m]      = make_float2(px, pz);
  Pb[base + 16 + m] = make_float2(py, px*px + py*py + pz*pz);

  float gx = R1[0]*x + R1[1]*y + R1[2]*z + t1[0];
  float gy = R1[3]*x + R1[4]*y + R1[5]*z + t1[1];
  float gz = R1[6]*x + R1[7]*y + R1[8]*z + t1[2];
  Gb[base + m]      = make_float2(gx, gz);
  Gb[base + 16 + m] = make_float2(gy, gx*gx + gy*gy + gz*gz);
}

// One wave owns a 16-row query tile and sweeps 256 target tiles with
// V_WMMA_F32_16X16X4_F32:
//   A (16x4): row r = (-2qx, -2qy, -2qz, 1)      E = A*B = |t|^2 - 2 q.t
//   B (4x16): col c = ( tx,  ty,  tz, |t|^2)     (loaded pre-swizzled)
// rowmin(E) + |q|^2, clamp, sqrt, sum over 16 rows -> one partial.
// gridDim.y: pass 0 = pred->gt (row mins), pass 1 = gt->pred (col mins).
__global__ __launch_bounds__(256)
void chamfer_pass(const float2* __restrict__ Pb, const float2* __restrict__ Gb,
                  float* __restrict__ partials) {
  const int pass = blockIdx.y;
  const float2* __restrict__ Qb = pass ? Gb : Pb;
  const float2* __restrict__ Tb = pass ? Pb : Gb;

  const int lane = threadIdx.x & 31;
  const int w    = blockIdx.x * 8 + (threadIdx.x >> 5);  // 0..2047
  const int b    = w >> 8;
  const int rt   = w & 255;
  const int h    = lane >> 4;
  const int m    = lane & 15;

  // A regs: V0 = K{0|2} (-2x | -2z), V1 = K{1|3} (-2y | 1) of row m
  const float2* qbase = Qb + ((b << 8) + rt) * 32;
  float2 q0 = qbase[m];        // (x_m, z_m)
  float2 q1 = qbase[16 + m];   // (y_m, |q_m|^2)
  v2f a;
  a.x = -2.0f * (h ? q0.y : q0.x);
  a.y = h ? 1.0f : (-2.0f * q1.x);
  const float nrmSelf = q1.y;  // lanes 0-15 hold |q_row|^2 for rows 0-15

  v8f mins;
  #pragma unroll
  for (int v = 0; v < 8; ++v) mins[v] = 3.0e38f;

  const float2* tbase = Tb + (b << 8) * 32;
  for (int ct0 = 0; ct0 < 256; ct0 += 4) {
    int pf = (ct0 + 32 < 256) ? (ct0 + 32) : 252;   // branchless clamp
    __builtin_prefetch(&tbase[pf*32 + lane], 0, 3);
    #pragma unroll
    for (int u = 0; u < 4; ++u) {
      float2 t2 = tbase[(ct0 + u)*32 + lane];       // B regs, pre-swizzled
      v2f bv; bv.x = t2.x; bv.y = t2.y;
      v8f c = {};
      v8f d = __builtin_amdgcn_wmma_f32_16x16x4_f32(
          /*neg_a=*/false, a, /*neg_b=*/false, bv,
          /*c_mod=*/(short)0, c, /*reuse_a=*/false, /*reuse_b=*/false);
      #pragma unroll
      for (int v = 0; v < 8; ++v) mins[v] = fminf(mins[v], d[v]);
    }
  }

  // min across the 16 lanes of each half (each half = one row group)
  #pragma unroll
  for (int v = 0; v < 8; ++v) {
    float x = mins[v];
    x = fminf(x, __shfl_xor(x, 1, 32));
    x = fminf(x, __shfl_xor(x, 2, 32));
    x = fminf(x, __shfl_xor(x, 4, 32));
    x = fminf(x, __shfl_xor(x, 8, 32));
    mins[v] = x;
  }

  // add per-row |q|^2 bias (lane-indexed shuffle), clamp, sqrt, accumulate
  float rowSum = 0.f;
  #pragma unroll
  for (int v = 0; v < 8; ++v) {
    int row = v + 8*h;
    float nrm = __shfl(nrmSelf, row, 32);
    rowSum += sqrtf(fmaxf(mins[v] + nrm, 0.0f));
  }
  rowSum += __shfl_xor(rowSum, 16, 32);
  if (lane == 0) partials[pass*2048 + w] = rowSum;
}

// Single wave, fixed-order (deterministic) reduction.
__global__ void finalize_kernel(const float* __restrict__ ws, float* __restrict__ out) {
  int l = threadIdx.x;                // 0..31
  const float* partials = ws + WS_PART;
  float s = 0.f;
  for (int i = l; i < 4096; i += 32) s += partials[i];
  #pragma unroll
  for (int msk = 16; msk >= 1; msk >>= 1) s += __shfl_xor(s, msk, 32);
  if (l == 0) {
    const float* tp = ws + WS_RELP_T;
    const float* tg = ws + WS_RELG_T;
    float lt = 0.f;
    for (int i = 0; i < 24; ++i) lt += fabsf(tp[i] - tg[i]);
    out[0] = s * (1.0f / 32768.0f);   // mean over N per batch, mean over B
    out[1] = lt * (1.0f / 24.0f);
  }
}

extern "C" void kernel_launch(void* const* d_in, const int* in_sizes, int n_in,
                              void* d_out, int out_size, void* d_ws, size_t ws_size,
                              hipStream_t stream) {
  const float* pred_rot  = (const float*)d_in[0];
  const float* pred_tr   = (const float*)d_in[1];
  const float* ctxh_rot  = (const float*)d_in[2];
  const float* ctxh_tr   = (const float*)d_in[3];
  const float* gt_rot    = (const float*)d_in[4];
  const float* gt_tr     = (const float*)d_in[5];
  const float* ctxg_rot  = (const float*)d_in[6];
  const float* ctxg_tr   = (const float*)d_in[7];
  const float* model_pts = (const float*)d_in[8];

  float*  w  = (float*)d_ws;
  float2* Pb = (float2*)(w + WS_PB);
  float2* Gb = (float2*)(w + WS_GB);

  setup_kernel<<<1, 32, 0, stream>>>(pred_rot, pred_tr, ctxh_rot, ctxh_tr,
                                     gt_rot, gt_tr, ctxg_rot, ctxg_tr, w);
  transform_kernel<<<128, 256, 0, stream>>>(model_pts, w, Pb, Gb);
  chamfer_pass<<<dim3(256, 2), 256, 0, stream>>>(Pb, Gb, w + WS_PART);
  finalize_kernel<<<1, 32, 0, stream>>>(w, (float*)d_out);
}